// MaxRelativeGraphConv_12429635355031
// MI455X (gfx1250) — compile-verified
//
#include <hip/hip_runtime.h>

#define IN_DIM  128
#define OUT_DIM 128
#define KEY_NEGINF 0x007FFFFFu   // monotonic key of -inf (bits 0xFF800000 -> ~u)

typedef __attribute__((ext_vector_type(2))) float v2f;
typedef __attribute__((ext_vector_type(8))) float v8f;

// Order-preserving float <-> uint mapping so unsigned atomicMax == float max.
__device__ __forceinline__ unsigned f2key(float f) {
  unsigned u = __float_as_uint(f);
  return (u & 0x80000000u) ? ~u : (u | 0x80000000u);
}
__device__ __forceinline__ float key2f(unsigned k) {
  unsigned u = (k & 0x80000000u) ? (k & 0x7FFFFFFFu) : ~k;
  return __uint_as_float(u);
}

// ---------------- Kernel 1: init max-keys to encode(-inf) ----------------
__global__ void mrconv_init_keys(uint4* __restrict__ keys4, int total4) {
  int i = blockIdx.x * blockDim.x + threadIdx.x;
  if (i < total4) {
    keys4[i] = make_uint4(KEY_NEGINF, KEY_NEGINF, KEY_NEGINF, KEY_NEGINF);
  }
}

// ---------------- Kernel 2: per-edge scatter-max of (x[src]-x[dst]) -------
// One wave32 per edge; each lane owns 4 contiguous channels (float4).
// src/dst are wave-uniform -> readfirstlane puts them in SGPRs so the x
// gathers / atomic bases use saddr addressing.
__global__ void mrconv_edge_max(const float* __restrict__ x,
                                const int* __restrict__ src,
                                const int* __restrict__ dst,
                                unsigned* __restrict__ keys, int E) {
  int wave = (blockIdx.x * blockDim.x + threadIdx.x) >> 5;
  int lane = threadIdx.x & 31;
  if (wave >= E) return;
  int s = __builtin_amdgcn_readfirstlane(src[wave]);
  int d = __builtin_amdgcn_readfirstlane(dst[wave]);
  const float4 xs = ((const float4*)(x + (size_t)s * IN_DIM))[lane];
  const float4 xd = ((const float4*)(x + (size_t)d * IN_DIM))[lane];
  unsigned* kd = keys + (size_t)d * IN_DIM + lane * 4;
  atomicMax(&kd[0], f2key(xs.x - xd.x));
  atomicMax(&kd[1], f2key(xs.y - xd.y));
  atomicMax(&kd[2], f2key(xs.z - xd.z));
  atomicMax(&kd[3], f2key(xs.w - xd.w));
}

// ---------------- Kernel 3: fused decode + [x | max_diff] @ W + b ---------
// Block = 256 threads = 8 waves. Block owns 16 output rows; wave w owns
// output columns [16w, 16w+16). A tile (16 x 256) staged once in LDS and
// shared by all 8 waves. fp32 WMMA 16x16x4, K = 256 -> 64 WMMAs per wave.
#define A_LDS_STRIDE 260   // 260 % 64 == 4 -> conflict-free strided reads
                           // row byte base = r*1040, 16B aligned -> b128 ok

__global__ void mrconv_gemm(const float* __restrict__ x,
                            const unsigned* __restrict__ keys,
                            const float* __restrict__ W,
                            const float* __restrict__ bias,
                            float* __restrict__ out, int N) {
  __shared__ float Atile[16 * A_LDS_STRIDE];

  const int tid = threadIdx.x;
  const int m0  = blockIdx.x * 16;

  // ---- Stage A tile with b128 accesses: 16 rows x 32 float4 per half ----
  // cols 0..127 from x; cols 128..255 decoded from max keys.
  {
    const float4* xsrc = (const float4*)x;
    #pragma unroll
    for (int j = 0; j < 2; ++j) {
      int e    = tid + j * 256;     // 0..511
      int row  = e >> 5;            // 0..15
      int c4   = e & 31;            // float4 index within row
      int grow = m0 + row;
      float4 v = make_float4(0.f, 0.f, 0.f, 0.f);
      if (grow < N) v = xsrc[(size_t)grow * (IN_DIM / 4) + c4];
      *(float4*)&Atile[row * A_LDS_STRIDE + c4 * 4] = v;
    }
    const uint4* ksrc = (const uint4*)keys;
    #pragma unroll
    for (int j = 0; j < 2; ++j) {
      int e    = tid + j * 256;
      int row  = e >> 5;
      int c4   = e & 31;
      int grow = m0 + row;
      float4 v = make_float4(0.f, 0.f, 0.f, 0.f);
      if (grow < N) {
        uint4 k = ksrc[(size_t)grow * (IN_DIM / 4) + c4];
        v.x = (k.x == KEY_NEGINF) ? 0.f : key2f(k.x);   // isolated nodes -> 0
        v.y = (k.y == KEY_NEGINF) ? 0.f : key2f(k.y);
        v.z = (k.z == KEY_NEGINF) ? 0.f : key2f(k.z);
        v.w = (k.w == KEY_NEGINF) ? 0.f : key2f(k.w);
      }
      *(float4*)&Atile[row * A_LDS_STRIDE + IN_DIM + c4 * 4] = v;
    }
  }
  __syncthreads();

  const int waveId = tid >> 5;        // 0..7  -> column tile
  const int lane   = tid & 31;
  const int half   = lane >> 4;       // 0: lanes 0-15, 1: lanes 16-31
  const int lr     = lane & 15;
  const int n0     = waveId * 16;

  v8f acc = {};

  // D = A(16x256) * W(256x16-slice), K stepped by 4 with f32 WMMA.
  // f32 A layout: lane lr holds A[lr][k+2h], A[lr][k+2h+1]
  // f32 B layout: lane lr holds W[k+2h][n0+lr], W[k+2h+1][n0+lr]
  const float* Arow = &Atile[lr * A_LDS_STRIDE + 2 * half];
  const float* Wcol = W + (size_t)2 * half * OUT_DIM + n0 + lr;
  #pragma unroll 16
  for (int kk = 0; kk < 2 * IN_DIM; kk += 4) {
    v2f a;
    a.x = Arow[kk];
    a.y = Arow[kk + 1];
    v2f bb;
    bb.x = Wcol[(size_t)kk * OUT_DIM];
    bb.y = Wcol[(size_t)(kk + 1) * OUT_DIM];
    acc = __builtin_amdgcn_wmma_f32_16x16x4_f32(
        /*neg_a=*/false, a, /*neg_b=*/false, bb,
        /*c_mod=*/(short)0, acc, /*reuse_a=*/false, /*reuse_b=*/false);
  }

  // D layout: VGPR i -> row (m0 + i + 8*half), col (n0 + lr).
  const float bv = bias[n0 + lr];
  if (m0 + 16 <= N) {
    // Fast path: straight-line stores off a common base (saddr + offset).
    float* op = out + (size_t)(m0 + 8 * half) * OUT_DIM + n0 + lr;
    #pragma unroll
    for (int i = 0; i < 8; ++i) {
      op[(size_t)i * OUT_DIM] = acc[i] + bv;
    }
  } else {
    #pragma unroll
    for (int i = 0; i < 8; ++i) {
      int row = m0 + i + 8 * half;
      if (row < N) out[(size_t)row * OUT_DIM + n0 + lr] = acc[i] + bv;
    }
  }
}

extern "C" void kernel_launch(void* const* d_in, const int* in_sizes, int n_in,
                              void* d_out, int out_size, void* d_ws, size_t ws_size,
                              hipStream_t stream) {
  const float*    x    = (const float*)d_in[0];
  const int*      ei   = (const int*)d_in[1];   // (2, E): row0 = src, row1 = dst
  const float*    W    = (const float*)d_in[2];
  const float*    bias = (const float*)d_in[3];
  float*          out  = (float*)d_out;
  unsigned*       keys = (unsigned*)d_ws;       // N * 128 uint32 = 25.6 MB

  const int N = in_sizes[0] / IN_DIM;
  const int E = in_sizes[1] / 2;

  // 1) init max keys to encode(-inf)
  {
    int total4 = (N * IN_DIM) / 4;
    int blocks = (total4 + 255) / 256;
    mrconv_init_keys<<<blocks, 256, 0, stream>>>((uint4*)keys, total4);
  }
  // 2) edge scatter-max: one wave per edge, 8 edges per 256-thread block
  {
    int blocks = (E + 7) / 8;
    mrconv_edge_max<<<blocks, 256, 0, stream>>>(x, ei, ei + E, keys, E);
  }
  // 3) fused decode + concat + GEMM + bias via fp32 WMMA
  {
    int blocks = (N + 15) / 16;
    mrconv_gemm<<<blocks, 256, 0, stream>>>(x, keys, W, bias, out, N);
  }
}